// Hypernetwork_PONG_Bayes_55181739819747
// MI455X (gfx1250) — compile-verified
//
#include <hip/hip_runtime.h>

typedef __attribute__((ext_vector_type(2))) float v2f;
typedef __attribute__((ext_vector_type(8))) float v8f;

// ---------------------------------------------------------------------------
// Split-K GEMV partial: out_partial[chunk][j] = sum_{i in chunk} x[i]*W[i,j]
// W row-major (rows x nCols). Coalesced: lane j reads consecutive addresses.
// ---------------------------------------------------------------------------
__global__ void gemv_partial(const float* __restrict__ x,
                             const float* __restrict__ W,
                             float* __restrict__ partial,
                             int nCols, int rowsPerChunk) {
  int j = blockIdx.x * blockDim.x + threadIdx.x;
  if (j >= nCols) return;
  int i0 = blockIdx.y * rowsPerChunk;
  const float* wp = W + (size_t)i0 * nCols + j;
  float acc = 0.f;
  for (int r = 0; r < rowsPerChunk; ++r) {
    acc = fmaf(x[i0 + r], *wp, acc);
    wp += nCols;
  }
  partial[(size_t)blockIdx.y * nCols + j] = acc;
}

// Deterministic fixed-order reduction of P partial rows + bias, then ReLU.
__global__ void reduce_bias_relu(const float* __restrict__ partial,
                                 const float* __restrict__ bias,
                                 float* __restrict__ out, int N, int P) {
  int j = blockIdx.x * blockDim.x + threadIdx.x;
  if (j >= N) return;
  float s = bias[j];
  for (int p = 0; p < P; ++p) s += partial[(size_t)p * N + j];
  out[j] = s > 0.f ? s : 0.f;
}

// ---------------------------------------------------------------------------
// Hidden layer of one hyper-segment, zero-padded to nPad rows:
//   H[r,k] = (r < n) ? relu(x2[base + 15r + 1] * Wa[k] + ba[k]) : 0
// ---------------------------------------------------------------------------
__global__ void seg_hidden(const float* __restrict__ x2, int base,
                           int n, int nPad, int h,
                           const float* __restrict__ Wa,
                           const float* __restrict__ ba,
                           float* __restrict__ H) {
  int idx = blockIdx.x * blockDim.x + threadIdx.x;
  if (idx >= nPad * h) return;
  int r = idx / h;
  int k = idx - r * h;
  float v = 0.f;
  if (r < n) {
    float s = x2[base + 15 * r + 1];
    v = fmaf(s, Wa[k], ba[k]);
    v = v > 0.f ? v : 0.f;
  }
  H[idx] = v;
}

// ---------------------------------------------------------------------------
// WMMA fp32 GEMM: out = relu(H(nPad x HD) @ Wb(HD x OD) + bb), one wave (32
// threads) per 16x16 output tile, K in steps of 4 via v_wmma_f32_16x16x4_f32.
// HD (=K) and OD (=N) are compile-time so B strides fold into load immediates
// and the store addressing is pure constant-stride pointer bumps.
//
// ISA layout (CDNA5 ISA):
//   A 16x4 f32: lanes 0-15 hold M=lane, K={0,1} in v0/v1; lanes 16-31 hold
//               M=lane-16, K={2,3}.
//   B 4x16 f32: symmetric (row K striped across lanes within a VGPR).
//   C/D 16x16:  lane half selects M-half, VGPR v = M row (v / v+8), N = lane&15.
//
// Column overrun: B loads use a clamped in-bounds address UNCONDITIONALLY.
// A lane's B value only feeds D-column N = lane&15, which is held by that
// same lane and never stored when colOK is false — so the garbage (finite)
// value is harmless and no per-iteration masking/EXEC divergence is needed.
// ---------------------------------------------------------------------------
template <int HD, int OD>
__global__ void seg_gemm_wmma(const float* __restrict__ H,
                              const float* __restrict__ Wb,
                              const float* __restrict__ bb,
                              float* __restrict__ out,
                              int n, int colTiles) {
  int tile = blockIdx.x;
  int rt = tile / colTiles;
  int ct = tile - rt * colTiles;
  int lane = threadIdx.x;        // blockDim.x == 32: exactly one wave32
  int m    = lane & 15;
  int half = lane >> 4;
  int kb   = half * 2;           // K sub-offset this lane owns
  int col  = ct * 16 + m;
  bool colOK = col < OD;
  int csafe  = colOK ? col : (OD - 1);   // always in-bounds

  const float* ap  = H + (size_t)(rt * 16 + m) * HD + kb;   // rows are padded
  const float* bp0 = Wb + (size_t)kb * OD + csafe;

  v8f acc = {};
#pragma unroll 5
  for (int k = 0; k < HD; k += 4) {
    v2f a = *(const v2f*)(ap + k);          // K = k+kb, k+kb+1 (8B aligned)
    v2f b;
    b.x = bp0[(size_t)k * OD];              // K = k+kb,   N = col (clamped)
    b.y = bp0[(size_t)k * OD + OD];         // K = k+kb+1, N = col (clamped)
    acc = __builtin_amdgcn_wmma_f32_16x16x4_f32(
        /*neg_a=*/false, a, /*neg_b=*/false, b,
        /*c_mod=*/(short)0, acc, /*reuse_a=*/false, /*reuse_b=*/false);
  }

  float bias = bb[csafe];
  int rbase = rt * 16 + half * 8;                 // first output row this lane holds
  float* op = out + (size_t)rbase * OD + col;     // constant stride OD between rows
  if (colOK) {
    if (rbase + 8 <= n) {                         // full tile: no row guards
#pragma unroll
      for (int v = 0; v < 8; ++v) {
        float val = acc[v] + bias;
        *op = val > 0.f ? val : 0.f;
        op += OD;
      }
    } else {                                      // ragged bottom tile (seg4)
#pragma unroll
      for (int v = 0; v < 8; ++v) {
        if (rbase + v < n) {
          float val = acc[v] + bias;
          *op = val > 0.f ? val : 0.f;
        }
        op += OD;
      }
    }
  }
}

// ---------------------------------------------------------------------------
extern "C" void kernel_launch(void* const* d_in, const int* in_sizes, int n_in,
                              void* d_out, int out_size, void* d_ws, size_t ws_size,
                              hipStream_t stream) {
  const float* inputs = (const float*)d_in[0];
  const float* Wd1 = (const float*)d_in[1];
  const float* bd1 = (const float*)d_in[2];
  const float* Wd2 = (const float*)d_in[3];
  const float* bd2 = (const float*)d_in[4];
  const float* W1a = (const float*)d_in[5];
  const float* b1a = (const float*)d_in[6];
  const float* W1b = (const float*)d_in[7];
  const float* b1b = (const float*)d_in[8];
  const float* W2a = (const float*)d_in[9];
  const float* b2a = (const float*)d_in[10];
  const float* W2b = (const float*)d_in[11];
  const float* b2b = (const float*)d_in[12];
  const float* W3a = (const float*)d_in[13];
  const float* b3a = (const float*)d_in[14];
  const float* W3b = (const float*)d_in[15];
  const float* b3b = (const float*)d_in[16];
  const float* W4a = (const float*)d_in[17];
  const float* b4a = (const float*)d_in[18];
  const float* W4b = (const float*)d_in[19];
  const float* b4b = (const float*)d_in[20];
  float* out = (float*)d_out;
  float* ws = (float*)d_ws;

  // Workspace layout (float offsets, all multiples of 4 -> 16B aligned)
  float* pA = ws + 0;        // 96 x 1000   = 96000
  float* pB = ws + 96000;    // 10 x 4605   = 46050
  float* y1 = ws + 142080;   // 1000
  float* x2 = ws + 143080;   // 4605
  float* H1 = ws + 147712;   // 16  x 500   = 8000
  float* H2 = ws + 155712;   // 32  x 100   = 3200
  float* H3 = ws + 158912;   // 256 x 100   = 25600
  float* H4 = ws + 184512;   // 16  x 100   = 1600  (3 valid rows, padded)

  // ---- Layer 1: y1 = relu(inputs @ Wd1 + bd1), split-K 96 x 350 rows ----
  gemv_partial<<<dim3(4, 96), 256, 0, stream>>>(inputs, Wd1, pA, 1000, 350);
  reduce_bias_relu<<<4, 256, 0, stream>>>(pA, bd1, y1, 1000, 96);

  // ---- Layer 2: x2 = relu(y1 @ Wd2 + bd2), split-K 10 x 100 rows ----
  gemv_partial<<<dim3(18, 10), 256, 0, stream>>>(y1, Wd2, pB, 4605, 100);
  reduce_bias_relu<<<18, 256, 0, stream>>>(pB, bd2, x2, 4605, 10);

  // ---- Hidden activations per segment (zero-padded rows) ----
  seg_hidden<<<32, 256, 0, stream>>>(x2,    0,  16,  16, 500, W1a, b1a, H1);
  seg_hidden<<<13, 256, 0, stream>>>(x2,  240,  32,  32, 100, W2a, b2a, H2);
  seg_hidden<<<100, 256, 0, stream>>>(x2, 720, 256, 256, 100, W3a, b3a, H3);
  seg_hidden<<<7, 256, 0, stream>>>(x2, 4560,   3,  16, 100, W4a, b4a, H4);

  // ---- Segment GEMMs via fp32 WMMA; outputs concatenated flat ----
  // seg1: 16x500 @ 500x5121 -> out[0..81936)
  seg_gemm_wmma<500, 5121><<<1 * 321, 32, 0, stream>>>(H1, W1b, b1b,
                                                       out + 0, 16, 321);
  // seg2: 32x100 @ 100x257 -> out[81936..90160)
  seg_gemm_wmma<100, 257><<<2 * 17, 32, 0, stream>>>(H2, W2b, b2b,
                                                     out + 81936, 32, 17);
  // seg3: 256x100 @ 100x321 -> out[90160..172336)
  seg_gemm_wmma<100, 321><<<16 * 21, 32, 0, stream>>>(H3, W3b, b3b,
                                                      out + 90160, 256, 21);
  // seg4: 3x100 @ 100x1542 -> out[172336..176962)
  seg_gemm_wmma<100, 1542><<<1 * 97, 32, 0, stream>>>(H4, W4b, b4b,
                                                      out + 172336, 3, 97);
}